// Quantizer_4157528342986
// MI455X (gfx1250) — compile-verified
//
#include <hip/hip_runtime.h>
#include <hip/hip_bf16.h>

typedef __attribute__((ext_vector_type(2))) float v2f;
typedef __attribute__((ext_vector_type(8))) float v8f;

#define N_GROUPS 4
#define N_CODES  160
#define GROUP_D  128
#define EMBED_C  512
#define T_LEN    4096
#define B_SZ     16
#define CB_STRIDE 132               // padded LDS row stride (floats) -> conflict-free A loads
#define ZQ_ELEMS ((size_t)B_SZ * EMBED_C * T_LEN)   // 33554432
#define LOSS_IDX ZQ_ELEMS
#define CODES_OFF (ZQ_ELEMS + 1)

__global__ __launch_bounds__(256) void vq_main(const float* __restrict__ xin,
                                               const float* __restrict__ cb,
                                               float* __restrict__ out,
                                               float* __restrict__ partial) {
    extern __shared__ float smem[];
    float* cbs = smem;                       // [160][132]
    float* c2s = smem + N_CODES * CB_STRIDE; // [160]  ||c_k||^2
    float* wls = c2s + N_CODES;              // [8] per-wave loss

    const int tid  = threadIdx.x;
    const int g    = blockIdx.z;
    const int b    = blockIdx.y;
    const int tblk = blockIdx.x * 256;
    const int lane = tid & 31;
    const int w    = tid >> 5;
    const int n    = lane & 15;   // vector (time) slot within tile
    const int hi   = lane >> 4;   // which K-half this lane carries

    // ---- stage this group's codebook into LDS (padded rows) ----
    const float* cbg = cb + (size_t)g * (N_CODES * GROUP_D);
    for (int i = tid; i < N_CODES * GROUP_D; i += 256) {
        int k = i >> 7, d = i & 127;
        cbs[k * CB_STRIDE + d] = cbg[i];
    }
    __syncthreads();
    for (int k = tid; k < N_CODES; k += 256) {
        float s = 0.f;
        #pragma unroll 4
        for (int d = 0; d < GROUP_D; ++d) { float v = cbs[k * CB_STRIDE + d]; s += v * v; }
        c2s[k] = s;
    }
    __syncthreads();

    float lossacc = 0.f;

    #pragma unroll 1
    for (int nt = 0; nt < 2; ++nt) {
        const int t0 = tblk + w * 32 + nt * 16;

        // ---- load x tile directly into WMMA B-operand layout ----
        // B (4x16): lanes 0-15 hold K=k0,k0+1 ; lanes 16-31 hold K=k0+2,k0+3
        const float* xp = xin + ((size_t)(b * EMBED_C + g * GROUP_D + 2 * hi)) * T_LEN + t0 + n;
        float xv[64];
        float x2 = 0.f;
        #pragma unroll
        for (int j = 0; j < 32; ++j) {
            float a0 = __builtin_nontemporal_load(xp + (size_t)(4 * j) * T_LEN);
            float a1 = __builtin_nontemporal_load(xp + (size_t)(4 * j + 1) * T_LEN);
            xv[2 * j] = a0; xv[2 * j + 1] = a1;
            x2 += a0 * a0 + a1 * a1;
        }

        float bestv = -3.4e38f;
        int   bestk = 0;

        #pragma unroll 1
        for (int kt = 0; kt < 10; ++kt) {
            const int kb = kt * 16;
            // init accumulator with -0.5*||c||^2 : score = x.c - 0.5||c||^2 (argmax == argmin dist)
            v8f acc;
            #pragma unroll
            for (int j2 = 0; j2 < 8; ++j2) acc[j2] = -0.5f * c2s[kb + j2 + 8 * hi];

            // A (16x4): lanes 0-15 rows M with K=d0,d0+1 ; lanes 16-31 same rows, K=d0+2,d0+3
            const float* arow = cbs + (size_t)(kb + n) * CB_STRIDE + 2 * hi;
            #pragma unroll
            for (int j = 0; j < 32; ++j) {
                v2f av = *(const v2f*)(arow + 4 * j);   // ds_load_b64, bank-conflict-free
                v2f bv; bv.x = xv[2 * j]; bv.y = xv[2 * j + 1];
                acc = __builtin_amdgcn_wmma_f32_16x16x4_f32(
                        false, av, false, bv, (short)0, acc, false, false);
            }
            // D layout: VGPR j -> M=j (lanes 0-15) / M=j+8 (lanes 16-31), N=lane&15
            #pragma unroll
            for (int j2 = 0; j2 < 8; ++j2) {
                float v = acc[j2];
                int   kk = kb + j2 + 8 * hi;
                if (v > bestv) { bestv = v; bestk = kk; }
            }
        }

        // combine the two K-halves held by lane L and L^16
        float ov = __shfl_xor(bestv, 16);
        int   ok = __shfl_xor(bestk, 16);
        if (ov > bestv || (ov == bestv && ok < bestk)) { bestv = ov; bestk = ok; }
        float x2f = x2 + __shfl_xor(x2, 16);
        if (hi == 0) lossacc += x2f - 2.f * bestv;   // = ||x - c*||^2

        // ---- codes output (as float, per harness output dtype) ----
        if (hi == 0) {
            __builtin_nontemporal_store((float)bestk,
                out + CODES_OFF + ((size_t)(b * N_GROUPS + g)) * T_LEN + t0 + n);
        }

        // ---- z_q output: lane writes its vector's 64-d half, coalesced along t ----
        float* zb = out + ((size_t)(b * EMBED_C + g * GROUP_D + 64 * hi)) * T_LEN + t0 + n;
        const float* crow = cbs + (size_t)bestk * CB_STRIDE + 64 * hi;
        #pragma unroll
        for (int jj = 0; jj < 64; ++jj) {
            __builtin_nontemporal_store(crow[jj], zb + (size_t)jj * T_LEN);
        }
    }

    // ---- per-block loss partial (deterministic) ----
    #pragma unroll
    for (int off = 16; off >= 1; off >>= 1) lossacc += __shfl_xor(lossacc, off);
    if (lane == 0) wls[w] = lossacc;
    __syncthreads();
    if (tid == 0) {
        float s = 0.f;
        #pragma unroll
        for (int i = 0; i < 8; ++i) s += wls[i];
        partial[((size_t)blockIdx.z * gridDim.y + blockIdx.y) * gridDim.x + blockIdx.x] = s;
    }
}

__global__ __launch_bounds__(256) void vq_loss_reduce(const float* __restrict__ partial,
                                                      int nparts, float* __restrict__ out) {
    __shared__ float sh[256];
    float s = 0.f;
    for (int i = threadIdx.x; i < nparts; i += 256) s += partial[i];
    sh[threadIdx.x] = s;
    __syncthreads();
    for (int off = 128; off >= 1; off >>= 1) {
        if (threadIdx.x < off) sh[threadIdx.x] += sh[threadIdx.x + off];
        __syncthreads();
    }
    if (threadIdx.x == 0)
        out[LOSS_IDX] = sh[0] * (1.25f / (float)ZQ_ELEMS);
}

extern "C" void kernel_launch(void* const* d_in, const int* in_sizes, int n_in,
                              void* d_out, int out_size, void* d_ws, size_t ws_size,
                              hipStream_t stream) {
    const float* xin = (const float*)d_in[0];   // [16, 512, 4096] f32
    const float* cb  = (const float*)d_in[1];   // [4, 160, 128] f32
    float* out = (float*)d_out;
    float* partial = (float*)d_ws;              // 1024 floats of scratch

    dim3 grid(T_LEN / 256, B_SZ, N_GROUPS);     // 16 x 16 x 4 = 1024 blocks
    size_t shbytes = (size_t)(N_CODES * CB_STRIDE + N_CODES + 8) * sizeof(float); // ~85 KB
    vq_main<<<grid, 256, shbytes, stream>>>(xin, cb, out, partial);
    vq_loss_reduce<<<1, 256, 0, stream>>>(partial, 1024, out);
}